// Fold_55791625175251
// MI455X (gfx1250) — compile-verified
//
#include <hip/hip_runtime.h>

// col2im / fold: out[b,p,q] = sum_{i,j} x[b, 3i+j, p+1-i, q+1-j]  (zeros OOB)
// x: (16, 9, 512, 512) f32, out: (16, 1, 512, 512) f32.
// HBM-bound (168 MB @ 23.3 TB/s ~= 7.2 us floor; AI ~0.22 FLOP/B, zero reuse).
// v2: 4 outputs/thread, 9x global_load_async_to_lds_b128 per thread (ASYNCcnt,
// no VGPR writeback pressure), ds_load_b128 readback, float4 store. ~4x fewer
// instructions per byte than v1 so issue rate cannot co-bottleneck the stream.

#define FOLD_H    512
#define FOLD_W    512
#define FOLD_CH   9
#define FOLD_TPB  256
#define FOLD_VEC  4   // outputs per thread

__global__ __launch_bounds__(FOLD_TPB)
void Fold_55791625175251_kernel(const float* __restrict__ x,
                                float* __restrict__ out) {
    // One float4 LDS slot per (channel, thread): 9*256*16 = 36 KB.
    __shared__ float4 smem[FOLD_CH * FOLD_TPB];

    const int tid = threadIdx.x;
    const size_t o0 = (size_t)blockIdx.x * (FOLD_TPB * FOLD_VEC)
                    + (size_t)tid * FOLD_VEC;          // first output index
    const int q0  = (int)(o0 & (FOLD_W - 1));          // multiple of 4
    const int row = (int)(o0 >> 9);                    // b*512 + p
    const int p   = row & (FOLD_H - 1);
    const int b   = row >> 9;

    const float* xb = x + (size_t)b * FOLD_CH * FOLD_H * FOLD_W;

    // Low 32 bits of generic shared address == wave-relative LDS byte offset.
    const unsigned lds_base = (unsigned)(unsigned long long)smem;

    // Row shift validity + clamped row index per i.
    int  rc[3];
    bool rv[3];
#pragma unroll
    for (int i = 0; i < 3; ++i) {
        const int r = p + 1 - i;
        rv[i] = (r >= 0) && (r < FOLD_H);
        rc[i] = min(max(r, 0), FOLD_H - 1);
    }

    // Issue 9 async global->LDS b128 loads. Base element e = ch*H*W + rc*W +
    // (q0+1-j) is provably within [0, 9*H*W-4] for every (ch,j,rc,q0), so no
    // clamping is needed; stray reads land in adjacent rows/channels of the
    // same batch slab and are masked out at sum time.
#pragma unroll
    for (int i = 0; i < 3; ++i) {
#pragma unroll
        for (int j = 0; j < 3; ++j) {
            const int ch = i * 3 + j;
            const int e  = ch * (FOLD_H * FOLD_W) + rc[i] * FOLD_W + (q0 + 1 - j);
            const unsigned long long gaddr = (unsigned long long)(xb + e);
            const unsigned lds_off =
                lds_base + 16u * (unsigned)(ch * FOLD_TPB + tid);

            // GLOBAL_LOAD_ASYNC_TO_LDS_B128, GV mode (64-bit vaddr, saddr=off).
            asm volatile("global_load_async_to_lds_b128 %0, %1, off"
                         :
                         : "v"(lds_off), "v"(gaddr)
                         : "memory");
        }
    }

    // Each lane consumes only its own LDS slots -> per-wave wait, no barrier.
    asm volatile("s_wait_asynccnt 0" ::: "memory");

    float a0 = 0.0f, a1 = 0.0f, a2 = 0.0f, a3 = 0.0f;
#pragma unroll
    for (int i = 0; i < 3; ++i) {
#pragma unroll
        for (int j = 0; j < 3; ++j) {
            const int ch = i * 3 + j;
            const float4 v = smem[ch * FOLD_TPB + tid];  // ds_load_b128
            const bool r  = rv[i];
            // Column OOB exists only at the two row ends:
            //   element 0 of a j==2 tap when q0==0   (s = -1)
            //   element 3 of a j==0 tap when q0==508 (s = 512)
            const bool c0 = r && !((j == 2) && (q0 == 0));
            const bool c3 = r && !((j == 0) && (q0 == FOLD_W - FOLD_VEC));
            a0 += c0 ? v.x : 0.0f;
            a1 += r  ? v.y : 0.0f;
            a2 += r  ? v.z : 0.0f;
            a3 += c3 ? v.w : 0.0f;
        }
    }

    float4 o;
    o.x = a0; o.y = a1; o.z = a2; o.w = a3;
    *(float4*)(out + o0) = o;                            // 16B-aligned b128 store
}

extern "C" void kernel_launch(void* const* d_in, const int* in_sizes, int n_in,
                              void* d_out, int out_size, void* d_ws, size_t ws_size,
                              hipStream_t stream) {
    (void)in_sizes; (void)n_in; (void)d_ws; (void)ws_size; (void)out_size;
    const float* x = (const float*)d_in[0];
    float* out     = (float*)d_out;

    // 16*512*512 outputs, 1024 per block -> 4096 blocks of 8 wave32s.
    const int nblocks = (16 * FOLD_H * FOLD_W) / (FOLD_TPB * FOLD_VEC);
    Fold_55791625175251_kernel<<<nblocks, FOLD_TPB, 0, stream>>>(x, out);
}